// QuantumEmbedding_33174327394860
// MI455X (gfx1250) — compile-verified
//
#include <hip/hip_runtime.h>
#include <hip/hip_bf16.h>

typedef __attribute__((ext_vector_type(2))) float v2f;
typedef __attribute__((ext_vector_type(8))) float v8f;

#define NQ  8
#define DIM 256   // 2^NQ

struct cf { float x, y; };
__device__ __forceinline__ cf cmul(cf a, cf b) {
    return { a.x * b.x - a.y * b.y, a.x * b.y + a.y * b.x };
}

// CNOT cascade c=0..6 (control c, target c+1) == MSB-first prefix-XOR of the
// 8-bit basis index: amplitude at j moves to perm8(j).
__device__ __forceinline__ int perm8(int j) {
    int y = 0, run = 0;
#pragma unroll
    for (int k = 7; k >= 0; --k) { run ^= (j >> k) & 1; y |= run << k; }
    return y;
}

// Row r (0/1) of U = Rz(2hz) @ Ry(2hy) @ Rx(2hx); returns columns 0 and 1.
__device__ __forceinline__ void u3_row(float hx, float hy, float hz, int r,
                                       cf& c0, cf& c1) {
    float sx, cx, sy, cy, sz, cz;
    __sincosf(hx, &sx, &cx);
    __sincosf(hy, &sy, &cy);
    __sincosf(hz, &sz, &cz);
    float t0 = cy * cx, t1 = sy * sx, t2 = sy * cx, t3 = cy * sx;
    cf a, b, ph;
    if (r) { a = { t2, -t3 }; b = { t0, -t1 }; ph = { cz,  sz }; }
    else   { a = { t0,  t1 }; b = { -t2, -t3 }; ph = { cz, -sz }; }
    c0 = cmul(ph, a);
    c1 = cmul(ph, b);
}

#define LDS_FENCE() asm volatile("s_wait_dscnt 0x0" ::: "memory")

// State layout in LDS (per 16x16 complex matrix, 512 floats):
//   element (r,c):  re at [r*32 + c],  im at [r*32 + 16 + c]
__global__ __launch_bounds__(256) void qsim_wmma_kernel(
    const float* __restrict__ x,    // [M, 256] embeddings
    const float* __restrict__ qp,   // [256, 8, 3] params
    float* __restrict__ out,        // [M, 256] complex64 -> interleaved floats
    int M) {
    __shared__ float lds[8][2][2 * DIM];   // [wave][ S | T ]

    const int lane = threadIdx.x & 31;
    const int wave = threadIdx.x >> 5;
    const int m    = lane & 15;   // lane's matrix row (A side) == column n (B side)
    const int hi   = lane >> 4;   // lane-half: selects K pairs / rows m+8
    const int circ = blockIdx.x * 8 + wave;
    if (circ >= M) return;        // wave-uniform

    float* S = &lds[wave][0][0];
    float* T = &lds[wave][1][0];

    // state = |00000000>
#pragma unroll
    for (int k = 0; k < 16; ++k) S[k * 32 + lane] = 0.0f;
    if (lane == 0) S[0] = 1.0f;
    LDS_FENCE();

    // Fixed per-lane store offsets. D-layout: VGPR v holds element
    // (r = v + 8*hi, c = m). Final store folds the CNOT permutation.
    int stT[8], stS[8];
#pragma unroll
    for (int v = 0; v < 8; ++v) {
        int r = v + 8 * hi;
        stT[v] = r * 32 + m;                       // plain (row-planar) offset
        int j  = perm8(r * 16 + m);                // permuted element index
        stS[v] = (j >> 4) * 32 + (j & 15);
    }

    float e_next = x[(size_t)circ * DIM];          // software-pipelined broadcast

    for (int i = 0; i < DIM; ++i) {
        const float e = e_next;
        int inext = (i + 1 < DIM) ? (i + 1) : (DIM - 1);
        e_next = x[(size_t)circ * DIM + inext];

        const float* p = qp + i * (NQ * 3);

        // --- single-qubit gate rows this lane needs -------------------------
        // A = U0(x)U1(x)U2(x)U3 acts on r; B = U4(x)U5(x)U6(x)U7 acts on c.
        const int b3 = (m >> 3) & 1, b2 = (m >> 2) & 1, b1 = (m >> 1) & 1, b0 = m & 1;
        cf u0a, u0b, u1a, u1b, u2a, u2b, u3a, u3b;   // qubits 0..3
        cf w0a, w0b, w1a, w1b, w2a, w2b, w3a, w3b;   // qubits 4..7
        u3_row(0.5f * e * p[0],  0.5f * e * p[1],  0.5f * e * p[2],  b3, u0a, u0b);
        u3_row(0.5f * e * p[3],  0.5f * e * p[4],  0.5f * e * p[5],  b2, u1a, u1b);
        u3_row(0.5f * e * p[6],  0.5f * e * p[7],  0.5f * e * p[8],  b1, u2a, u2b);
        u3_row(0.5f * e * p[9],  0.5f * e * p[10], 0.5f * e * p[11], b0, u3a, u3b);
        u3_row(0.5f * e * p[12], 0.5f * e * p[13], 0.5f * e * p[14], b3, w0a, w0b);
        u3_row(0.5f * e * p[15], 0.5f * e * p[16], 0.5f * e * p[17], b2, w1a, w1b);
        u3_row(0.5f * e * p[18], 0.5f * e * p[19], 0.5f * e * p[20], b1, w2a, w2b);
        u3_row(0.5f * e * p[21], 0.5f * e * p[22], 0.5f * e * p[23], b0, w3a, w3b);

        // Kron halves of the needed rows (4 complex each)
        cf AH[4], AL[4], BH[4], BL[4];
        AH[0] = cmul(u0a, u1a); AH[1] = cmul(u0a, u1b);
        AH[2] = cmul(u0b, u1a); AH[3] = cmul(u0b, u1b);
        AL[0] = cmul(u2a, u3a); AL[1] = cmul(u2a, u3b);
        AL[2] = cmul(u2b, u3a); AL[3] = cmul(u2b, u3b);
        BH[0] = cmul(w0a, w1a); BH[1] = cmul(w0a, w1b);
        BH[2] = cmul(w0b, w1a); BH[3] = cmul(w0b, w1b);
        BL[0] = cmul(w2a, w3a); BL[1] = cmul(w2a, w3b);
        BL[2] = cmul(w2b, w3a); BL[3] = cmul(w2b, w3b);

        // Operand construction in native WMMA f32 16x16x4 layouts:
        //  A-operand slice g: lane holds (m, kk), kk = 4g + t + 2*hi, t=0,1
        //  B-operand slice g: lane holds (k, n=m), k  = 4g + t + 2*hi
        // Y = B^T, so Y(k,n) = B(n,k) -> same (row m, column kk) product form.
        v2f aR[4], aI[4], maI[4], yR[4], yI[4], myI[4];
#pragma unroll
        for (int g = 0; g < 4; ++g) {
#pragma unroll
            for (int t = 0; t < 2; ++t) {
                int kk = 4 * g + t + 2 * hi;
                cf av = cmul(AH[kk >> 2], AL[kk & 3]);   // A(m,kk)
                cf yv = cmul(BH[kk >> 2], BL[kk & 3]);   // B(m,kk) == Y(kk,m)
                aR[g][t] = av.x;  aI[g][t] = av.y;  maI[g][t] = -av.y;
                yR[g][t] = yv.x;  yI[g][t] = yv.y;  myI[g][t] = -yv.y;
            }
        }

        // --- load state S as B-operand slices (2addr-friendly) --------------
        v2f sR[4], sI[4];
#pragma unroll
        for (int g = 0; g < 4; ++g) {
            int k0 = 4 * g + 2 * hi;
            int a0 = k0 * 32 + m;        // re(k0,m);  re(k1,m) at +32
            sR[g][0] = S[a0];       sR[g][1] = S[a0 + 32];
            sI[g][0] = S[a0 + 16];  sI[g][1] = S[a0 + 48];
        }

        // --- T = A * S (complex) : 16 WMMAs ---------------------------------
        v8f tr = {}; v8f ti = {};
#pragma unroll
        for (int g = 0; g < 4; ++g)
            tr = __builtin_amdgcn_wmma_f32_16x16x4_f32(false, aR[g],  false, sR[g], (short)0, tr, false, false);
#pragma unroll
        for (int g = 0; g < 4; ++g)
            tr = __builtin_amdgcn_wmma_f32_16x16x4_f32(false, maI[g], false, sI[g], (short)0, tr, false, false);
#pragma unroll
        for (int g = 0; g < 4; ++g)
            ti = __builtin_amdgcn_wmma_f32_16x16x4_f32(false, aR[g],  false, sI[g], (short)0, ti, false, false);
#pragma unroll
        for (int g = 0; g < 4; ++g)
            ti = __builtin_amdgcn_wmma_f32_16x16x4_f32(false, aI[g],  false, sR[g], (short)0, ti, false, false);

        // --- stage T (D layout -> LDS, row-planar) --------------------------
#pragma unroll
        for (int v = 0; v < 8; ++v) {
            T[stT[v]]      = tr[v];
            T[stT[v] + 16] = ti[v];
        }
        LDS_FENCE();

        // --- reload T as A-operand: consecutive floats -> direct v2f b64 ----
        v2f txR[4], txI[4];
#pragma unroll
        for (int g = 0; g < 4; ++g) {
            int kk0 = 4 * g + 2 * hi;                    // even -> 8B aligned
            txR[g] = *(const v2f*)&T[m * 32 + kk0];
            txI[g] = *(const v2f*)&T[m * 32 + 16 + kk0];
        }

        // --- S' = T * B^T (complex) : 16 WMMAs ------------------------------
        v8f sr = {}; v8f si = {};
#pragma unroll
        for (int g = 0; g < 4; ++g)
            sr = __builtin_amdgcn_wmma_f32_16x16x4_f32(false, txR[g], false, yR[g],  (short)0, sr, false, false);
#pragma unroll
        for (int g = 0; g < 4; ++g)
            sr = __builtin_amdgcn_wmma_f32_16x16x4_f32(false, txI[g], false, myI[g], (short)0, sr, false, false);
#pragma unroll
        for (int g = 0; g < 4; ++g)
            si = __builtin_amdgcn_wmma_f32_16x16x4_f32(false, txR[g], false, yI[g],  (short)0, si, false, false);
#pragma unroll
        for (int g = 0; g < 4; ++g)
            si = __builtin_amdgcn_wmma_f32_16x16x4_f32(false, txI[g], false, yR[g],  (short)0, si, false, false);

        // --- store with CNOT-cascade permutation folded into the address ----
#pragma unroll
        for (int v = 0; v < 8; ++v) {
            S[stS[v]]      = sr[v];
            S[stS[v] + 16] = si[v];
        }
        LDS_FENCE();
    }

    // epilogue: convert row-planar -> interleaved complex64, coalesced b64
#pragma unroll
    for (int v = 0; v < 8; ++v) {
        int r = v + 8 * hi;
        v2f w;
        w[0] = S[r * 32 + m];
        w[1] = S[r * 32 + 16 + m];
        *(v2f*)&out[((size_t)circ * DIM + r * 16 + m) * 2] = w;
    }
}

extern "C" void kernel_launch(void* const* d_in, const int* in_sizes, int n_in,
                              void* d_out, int out_size, void* d_ws, size_t ws_size,
                              hipStream_t stream) {
    const float* x  = (const float*)d_in[0];   // [8,256,256] f32
    const float* qp = (const float*)d_in[1];   // [256,8,3]   f32
    float* out = (float*)d_out;                // [8,256,256] complex64 (re,im pairs)
    const int M = in_sizes[0] / DIM;           // 2048 circuits
    dim3 block(256);                           // 8 waves = 8 circuits / block
    dim3 grid((M + 7) / 8);
    qsim_wmma_kernel<<<grid, block, 0, stream>>>(x, qp, out, M);
}